// TopkActivation_78761110274618
// MI455X (gfx1250) — compile-verified
//
#include <hip/hip_runtime.h>
#include <stdint.h>

#ifndef __has_builtin
#define __has_builtin(x) 0
#endif

#define THREADS     256
#define H_DIM       32768
#define K_TOP       64
#define PER_THREAD  (H_DIM / THREADS)            // 128
#define CHUNKS      (H_DIM / (THREADS * 4))      // 32 x b128 per thread
// Pad LDS: +4 dwords per 128-dword segment -> segment walks are only 2-way
// bank conflicted, strided/chunked sweeps stay conflict-free & 16B aligned.
#define PAD_DWORDS  (H_DIM + 4 * (H_DIM / 128))  // 33792
#define SMEM_DWORDS (PAD_DWORDS + 256 + 256)     // + hist + scan

#if __has_builtin(__builtin_amdgcn_global_load_async_to_lds_b128)
#define HAVE_ASYNC_LD 1
#else
#define HAVE_ASYNC_LD 0
#endif
#if __has_builtin(__builtin_amdgcn_global_store_async_from_lds_b128)
#define HAVE_ASYNC_ST 1
#else
#define HAVE_ASYNC_ST 0
#endif

#if HAVE_ASYNC_LD || HAVE_ASYNC_ST
typedef int v4i __attribute__((ext_vector_type(4)));
typedef __attribute__((address_space(1))) v4i* g_v4i_p;  // global int4*
typedef __attribute__((address_space(3))) v4i* l_v4i_p;  // LDS int4*
__device__ __forceinline__ void wait_async0() {
#if __has_builtin(__builtin_amdgcn_s_wait_asynccnt)
  __builtin_amdgcn_s_wait_asynccnt(0);
#else
  asm volatile("s_wait_asynccnt 0" ::: "memory");
#endif
}
#endif

// padded LDS index
__device__ __forceinline__ int lds_phys(int e) { return e + ((e >> 7) << 2); }

// order-preserving float->uint key (desc order of float == desc order of key)
__device__ __forceinline__ unsigned key_of_bits(unsigned b) {
  return b ^ ((b & 0x80000000u) ? 0xFFFFFFFFu : 0x80000000u);
}
__device__ __forceinline__ float val_of_key(unsigned kv) {
  return __uint_as_float(kv ^ ((kv & 0x80000000u) ? 0x80000000u : 0xFFFFFFFFu));
}

__global__ void __launch_bounds__(THREADS)
TopkActivation_78761110274618_kernel(const float* __restrict__ x,
                                     float* __restrict__ out) {
  __shared__ __align__(16) unsigned smem[SMEM_DWORDS];
  unsigned* keys = smem;                 // padded row of keys / output bits
  unsigned* hist = smem + PAD_DWORDS;    // 256-bin histogram
  unsigned* scan = hist + 256;           // block scan scratch
  __shared__ unsigned s_selbin, s_selrem;

  const int tid = threadIdx.x;
  const size_t row = blockIdx.x;
  const float* __restrict__ xr = x + row * (size_t)H_DIM;
  float* __restrict__ outr = out + row * (size_t)H_DIM;

  // ---- Phase 1: HBM -> LDS (single read of the row) ----
#if HAVE_ASYNC_LD
  for (int c = 0; c < CHUNKS; ++c) {
    int e = (c * THREADS + tid) * 4;
    __builtin_amdgcn_global_load_async_to_lds_b128(
        (g_v4i_p)(const_cast<float*>(xr) + e),
        (l_v4i_p)&smem[lds_phys(e)], 0, 0);
  }
  wait_async0();
#else
  for (int c = 0; c < CHUNKS; ++c) {
    int e = (c * THREADS + tid) * 4;
    float4 v = *reinterpret_cast<const float4*>(xr + e);
    *reinterpret_cast<float4*>(&smem[lds_phys(e)]) = v;
  }
#endif
  __syncthreads();

  // ---- Phase 2: in-place transform to orderable keys ----
  for (int i = tid; i < H_DIM; i += THREADS) {
    int p = lds_phys(i);
    keys[p] = key_of_bits(keys[p]);
  }
  __syncthreads();

  // ---- Phase 3: radix-select the K-th largest key (4 x 8-bit rounds) ----
  unsigned prefix = 0;
  unsigned rem = K_TOP;  // rank remaining within current prefix class
  for (int shift = 24; shift >= 0; shift -= 8) {
    hist[tid] = 0;
    __syncthreads();
    unsigned pmask = (shift == 24) ? 0u : (0xFFFFFFFFu << (shift + 8));
    for (int i = tid; i < H_DIM; i += THREADS) {
      unsigned kv = keys[lds_phys(i)];
      if ((kv & pmask) == prefix)
        atomicAdd(&hist[(kv >> shift) & 0xFFu], 1u);
    }
    __syncthreads();
    // inclusive suffix sum over 256 bins (Hillis-Steele)
    unsigned mine = hist[tid];
    scan[tid] = mine;
    __syncthreads();
    for (int off = 1; off < 256; off <<= 1) {
      unsigned add = (tid + off < 256) ? scan[tid + off] : 0u;
      __syncthreads();
      scan[tid] += add;
      __syncthreads();
    }
    unsigned incl = scan[tid];          // count of elems with bin >= tid
    unsigned excl = incl - mine;        // count of elems with bin >  tid
    if (excl < rem && incl >= rem) {    // exactly one thread matches
      s_selbin = (unsigned)tid;
      s_selrem = rem - excl;
    }
    __syncthreads();
    prefix |= (s_selbin << shift);
    rem = s_selrem;
    __syncthreads();
  }
  const unsigned thresh = prefix;  // exact key of K-th largest
  const unsigned nties = rem;      // #elements == thresh to keep (index order)

  // ---- Phase 4: index-ordered tie resolution (matches lax.top_k stability) ----
  const int base = tid * PER_THREAD;  // contiguous logical segment per thread
  unsigned eq = 0;
  for (int i = 0; i < PER_THREAD; ++i)
    eq += (keys[lds_phys(base + i)] == thresh) ? 1u : 0u;
  scan[tid] = eq;
  __syncthreads();
  for (int off = 1; off < 256; off <<= 1) {  // inclusive ascending prefix sum
    unsigned add = (tid >= off) ? scan[tid - off] : 0u;
    __syncthreads();
    scan[tid] += add;
    __syncthreads();
  }
  unsigned ord = scan[tid] - eq;  // ordinal of my first tie, global index order

  // ---- Phase 5: write final output bits into LDS ----
  for (int i = 0; i < PER_THREAD; ++i) {
    int p = lds_phys(base + i);
    unsigned kv = keys[p];
    float o = 0.0f;
    if (kv > thresh) {
      o = val_of_key(kv);
    } else if (kv == thresh) {
      if (ord < nties) o = val_of_key(kv);
      ++ord;
    }
    keys[p] = __float_as_uint(o);
  }
  __syncthreads();

  // ---- Phase 6: LDS -> HBM (single write of the row) ----
#if HAVE_ASYNC_ST
  for (int c = 0; c < CHUNKS; ++c) {
    int e = (c * THREADS + tid) * 4;
    __builtin_amdgcn_global_store_async_from_lds_b128(
        (g_v4i_p)(outr + e), (l_v4i_p)&smem[lds_phys(e)], 0, 0);
  }
  wait_async0();
#else
  for (int c = 0; c < CHUNKS; ++c) {
    int e = (c * THREADS + tid) * 4;
    *reinterpret_cast<float4*>(outr + e) =
        *reinterpret_cast<const float4*>(&smem[lds_phys(e)]);
  }
#endif
}

extern "C" void kernel_launch(void* const* d_in, const int* in_sizes, int n_in,
                              void* d_out, int out_size, void* d_ws, size_t ws_size,
                              hipStream_t stream) {
  (void)in_sizes; (void)n_in; (void)d_ws; (void)ws_size;
  const float* x = (const float*)d_in[0];
  // d_in[1] holds k; fixed to 64 per the reference setup (K_TOP compile-time).
  float* out = (float*)d_out;
  const int B = out_size / H_DIM;  // 4096
  dim3 grid((unsigned)B), block(THREADS);
  hipLaunchKernelGGL(TopkActivation_78761110274618_kernel, grid, block, 0,
                     stream, x, out);
}